// RBFDescriptorFlax_74706661146715
// MI455X (gfx1250) — compile-verified
//
#include <hip/hip_runtime.h>
#include <stdint.h>

// ---------------------------------------------------------------------------
// RBF edge descriptor: gather-distance-exp-scatter.
//   out[a, b] = sum_{e : idx1[e]==a} exp(-beta * (shift_b - |R[idx0_e]-R[idx1_e]|)^2)
// Memory/atomic bound; no dense contraction -> no WMMA. CDNA5 paths used:
//   * global_load_async_to_lds (ASYNCcnt, double-buffered index staging)
//   * ds_bpermute_b32 wave32 broadcast
//   * line-coalesced no-return global_atomic_add_f32
//     (16 lanes own one edge's 16 consecutive dwords = one aligned 64B line)
// ---------------------------------------------------------------------------

#define BLOCK_THREADS   256
#define WAVES_PER_BLOCK (BLOCK_THREADS / 32)
#define BATCH           32          // edges per wave per iteration

#if defined(__has_builtin)
#  if __has_builtin(__builtin_amdgcn_global_load_async_to_lds_b32)
#    define RBF_ASYNC_LDS 1
#  endif
#  if __has_builtin(__builtin_amdgcn_s_wait_asynccnt)
#    define RBF_WAIT_BUILTIN 1
#  endif
#endif

typedef __attribute__((address_space(1))) int* rbf_g_i32;   // global int*
typedef __attribute__((address_space(3))) int* rbf_l_i32;   // LDS int*

template <int N>
__device__ __forceinline__ void rbf_wait_async() {
#if defined(RBF_ASYNC_LDS)
#  if defined(RBF_WAIT_BUILTIN)
    __builtin_amdgcn_s_wait_asynccnt(N);
#  else
    if constexpr (N == 0) { asm volatile("s_wait_asynccnt 0" ::: "memory"); }
    else                  { asm volatile("s_wait_asynccnt 2" ::: "memory"); }
#  endif
#endif
    asm volatile("" ::: "memory");   // compiler ordering fence for the LDS reads
}

// Stage the 32 indices of one batch into this wave's private LDS slice.
__device__ __forceinline__ void rbf_stage(const int* __restrict__ idx0,
                                          const int* __restrict__ idx1,
                                          int e, int lane,
                                          int* s0, int* s1) {
#if defined(RBF_ASYNC_LDS)
    __builtin_amdgcn_global_load_async_to_lds_b32(
        (rbf_g_i32)(uintptr_t)(idx0 + e),
        (rbf_l_i32)(uint32_t)(uintptr_t)(s0 + lane), 0, 0);
    __builtin_amdgcn_global_load_async_to_lds_b32(
        (rbf_g_i32)(uintptr_t)(idx1 + e),
        (rbf_l_i32)(uint32_t)(uintptr_t)(s1 + lane), 0, 0);
#else
    s0[lane] = idx0[e];
    s1[lane] = idx1[e];
    __builtin_prefetch(idx0 + e + 4096, 0, 0);   // global_prefetch_b8 on the stream
    __builtin_prefetch(idx1 + e + 4096, 0, 0);
#endif
}

__device__ __forceinline__ float rbf_bperm_f(int srcLane, float v) {
    return __builtin_bit_cast(float,
        __builtin_amdgcn_ds_bpermute(srcLane << 2, __builtin_bit_cast(int, v)));
}
__device__ __forceinline__ int rbf_bperm_i(int srcLane, int v) {
    return __builtin_amdgcn_ds_bpermute(srcLane << 2, v);
}

// Force the native no-return f32 atomic (global_atomic_add_f32), never a CAS loop.
__device__ __forceinline__ void rbf_atomic_add(float* p, float v) {
#if defined(__HIP_DEVICE_COMPILE__)
    unsafeAtomicAdd(p, v);
#else
    (void)p; (void)v;
#endif
}

__global__ void __launch_bounds__(BLOCK_THREADS)
rbf_zero_kernel(float* __restrict__ out, int n) {
    int i      = (int)(blockIdx.x * blockDim.x + threadIdx.x);
    int stride = (int)(gridDim.x * blockDim.x);
    for (; i < n; i += stride) out[i] = 0.0f;
}

__global__ void __launch_bounds__(BLOCK_THREADS)
rbf_scatter_kernel(const float* __restrict__ R,
                   const int*   __restrict__ idx0,
                   const int*   __restrict__ idx1,
                   float*       __restrict__ out,
                   int nEdges, int nBatches) {
    __shared__ int s0[WAVES_PER_BLOCK][2][BATCH];
    __shared__ int s1[WAVES_PER_BLOCK][2][BATCH];

    const int lane   = (int)(threadIdx.x & 31u);
    const int wslot  = (int)(threadIdx.x >> 5);
    const int gwave  = (int)blockIdx.x * WAVES_PER_BLOCK + wslot;
    const int nWaves = (int)gridDim.x * WAVES_PER_BLOCK;

    // betta = 16^2 / 6^2 ; fold log2(e) so exp() is a single v_exp_f32
    constexpr float kBetta = 256.0f / 36.0f;
    const float C2      = -(kBetta * 1.44269504088896f);
    const int   myB     = lane & 15;
    const float myShift = 0.5f + ((6.0f - 0.5f) / 16.0f) * (float)myB;

    int batch = gwave;
    if (batch >= nBatches) return;

    {   // prologue: stage first batch into buffer 0
        int e = batch * BATCH + lane;
        if (e >= nEdges) e = nEdges - 1;
        rbf_stage(idx0, idx1, e, lane, s0[wslot][0], s1[wslot][0]);
    }

    int k = 0;
    for (; batch < nBatches; batch += nWaves, ++k) {
        const int cur       = k & 1;
        const int nextBatch = batch + nWaves;

        if (nextBatch < nBatches) {   // software pipeline: stage k+1, wait for k
            int e = nextBatch * BATCH + lane;
            if (e >= nEdges) e = nEdges - 1;
            rbf_stage(idx0, idx1, e, lane, s0[wslot][cur ^ 1], s1[wslot][cur ^ 1]);
            rbf_wait_async<2>();      // in-order: the 2 oldest (batch k) retired
        } else {
            rbf_wait_async<0>();
        }

        // ---- phase 1: thread-per-edge distance (32 edges per wave) ----
        const int  myEdge = batch * BATCH + lane;
        const bool valid  = myEdge < nEdges;
        const volatile int* p0 = s0[wslot][cur];
        const volatile int* p1 = s1[wslot][cur];
        const int i0 = p0[lane];
        const int i1 = p1[lane];

        const float ax = R[i0 * 3 + 0], ay = R[i0 * 3 + 1], az = R[i0 * 3 + 2];
        const float bx = R[i1 * 3 + 0], by = R[i1 * 3 + 1], bz = R[i1 * 3 + 2];
        const float dx = ax - bx, dy = ay - by, dz = az - bz;
        const float d  = __builtin_amdgcn_sqrtf(fmaf(dx, dx, fmaf(dy, dy, dz * dz)));
        const int  key = valid ? i1 : -1;

        // ---- phase 2: 16 lanes per edge -> line-coalesced atomics ----
        // pass p handles edges 2p (lanes 0-15) and 2p+1 (lanes 16-31);
        // each atomic instruction touches 2 aligned 64B lines. Invalid (tail)
        // lanes contribute exactly 0.0f to out[0] instead of branching, so the
        // unrolled loop carries no per-pass EXEC manipulation.
        #pragma unroll
        for (int p = 0; p < 16; ++p) {
            const int   srcLane = 2 * p + (lane >> 4);
            const float de = rbf_bperm_f(srcLane, d);
            const int   ke = rbf_bperm_i(srcLane, key);
            const float t  = myShift - de;
            float       v  = __builtin_amdgcn_exp2f(C2 * (t * t));
            v = (ke >= 0) ? v : 0.0f;              // v_cndmask, no exec churn
            const int row = (ke >= 0) ? ke : 0;    // clamp for safe address
            rbf_atomic_add(out + (row * 16 + myB), v);
        }
    }
}

extern "C" void kernel_launch(void* const* d_in, const int* in_sizes, int n_in,
                              void* d_out, int out_size, void* d_ws, size_t ws_size,
                              hipStream_t stream) {
    (void)n_in; (void)d_ws; (void)ws_size;

    const float* R      = (const float*)d_in[0];       // [nAtoms, 3] f32
    const int*   idx    = (const int*)d_in[1];         // [2, nEdges] i32
    const int    nEdges = in_sizes[1] / 2;
    const int*   idx0   = idx;
    const int*   idx1   = idx + nEdges;
    float*       out    = (float*)d_out;               // [nAtoms, 16] f32

    // d_out is poisoned by the harness -> zero it every call.
    rbf_zero_kernel<<<dim3(1024), dim3(BLOCK_THREADS), 0, stream>>>(out, out_size);

    const int nBatches     = (nEdges + BATCH - 1) / BATCH;           // 200000
    int       blocksNeeded = (nBatches + WAVES_PER_BLOCK - 1) / WAVES_PER_BLOCK;
    const int blocks       = blocksNeeded < 4096 ? blocksNeeded : 4096;

    rbf_scatter_kernel<<<dim3(blocks), dim3(BLOCK_THREADS), 0, stream>>>(
        R, idx0, idx1, out, nEdges, nBatches);
}